// NodeProcessor_22840636080715
// MI455X (gfx1250) — compile-verified
//
#include <hip/hip_runtime.h>

#define N_NODE 50000
#define N_EDGE 625000
#define FDIM 128
#define HID 128
#define IN_DIM 256
#define N_TILES (N_NODE / 16)   // 3125, exact
#define WAVES_PER_BLOCK 4
#define LN_EPS 1e-5f

typedef float v2f __attribute__((ext_vector_type(2)));
typedef float v8f __attribute__((ext_vector_type(8)));

// ---------------------------------------------------------------------------
// Phase 0: zero the aggregation buffer in workspace (b128 stores)
// ---------------------------------------------------------------------------
__global__ __launch_bounds__(256) void zero_agg_kernel(float4* __restrict__ agg4) {
    size_t i = (size_t)blockIdx.x * blockDim.x + threadIdx.x;
    if (i < (size_t)N_NODE * FDIM / 4) agg4[i] = make_float4(0.f, 0.f, 0.f, 0.f);
}

// ---------------------------------------------------------------------------
// Phase 1: scatter-add edge features into receiver nodes.
// 32 threads/edge: one b128 load + 4 f32 atomics each; contiguous targets.
// agg (25.6MB) is L2-resident (192MB L2) so atomics resolve at L2.
// ---------------------------------------------------------------------------
__global__ __launch_bounds__(256) void scatter_add_kernel(
        const float4* __restrict__ edge_feature4,
        const int*    __restrict__ edge_matrix,
        float*        __restrict__ agg) {
    size_t gid = (size_t)blockIdx.x * blockDim.x + threadIdx.x;
    if (gid >= (size_t)N_EDGE * 32) return;
    const int e = (int)(gid >> 5);
    const int q = (int)(gid & 31);            // feature quad index
    const int r = edge_matrix[N_EDGE + e];    // row 1 of [2, N_EDGE]
    const float4 v = edge_feature4[(size_t)e * 32 + q];
    float* dst = agg + (size_t)r * FDIM + q * 4;
    atomicAdd(dst + 0, v.x);
    atomicAdd(dst + 1, v.y);
    atomicAdd(dst + 2, v.z);
    atomicAdd(dst + 3, v.w);
}

// ---------------------------------------------------------------------------
// Phase 2: fused MLP (256->128 relu, 128->128 relu, 128->128) + LayerNorm +
// residual via V_WMMA_F32_16X16X4_F32. One wave per 16-row node tile,
// 4 waves/block. Weights staged once per block into the 320KB WGP LDS in a
// ROW-PAIR INTERLEAVED, XOR-SWIZZLED layout:
//   sWp[j*256 + ((2c) ^ ((j&1)<<5))] = { W[2j][c], W[2j+1][c] }
// so each WMMA B fragment is ONE conflict-free aligned ds_load_b64 (the two
// half-waves use disjoint 32-bank halves), landing directly in a VGPR pair.
// ---------------------------------------------------------------------------
__device__ __forceinline__ int wpair_addr(int j, int c) {
    return j * 256 + (((c << 1) ^ ((j & 1) << 5)));
}

__global__ __launch_bounds__(32 * WAVES_PER_BLOCK) void mlp_ln_kernel(
        const float* __restrict__ nf,    // node_feature [N_NODE, 128]
        const float* __restrict__ agg,   // [N_NODE, 128]
        const float* __restrict__ W1,    // [256, 128]
        const float* __restrict__ b1,    // [128]
        const float* __restrict__ W2,    // [128, 128]
        const float* __restrict__ b2,    // [128]
        const float* __restrict__ W3,    // [128, 128]
        const float* __restrict__ b3,    // [128]
        const float* __restrict__ gamma, // [128]
        const float* __restrict__ beta,  // [128]
        float*       __restrict__ out) { // [N_NODE, 128]
    // Pair-interleaved weights: same byte totals as plain layout (256KB)
    __shared__ __align__(16) float sW1p[(IN_DIM / 2) * 256];  // 128KB
    __shared__ __align__(16) float sW2p[(HID / 2) * 256];     //  64KB
    __shared__ __align__(16) float sW3p[(HID / 2) * 256];     //  64KB
    // Per-wave activation buffer, padded to 130 dwords/row (130 mod 64 = 2)
    __shared__ __align__(16) float hbuf_all[WAVES_PER_BLOCK][16 * 130]; // 32.5KB

    const int tid  = threadIdx.x;
    const int wave = tid >> 5;
    const int lane = tid & 31;
    const int hh   = lane >> 4;      // half-wave: 0 or 1
    const int mcol = lane & 15;      // M (for A) / N (for B,C,D) index

    // ---- Cooperative weight staging into pair-interleaved layout ----------
    // i enumerates (row-pair j, column c); global reads coalesced per row,
    // LDS writes are aligned b64.
    {
        for (int i = tid; i < (IN_DIM / 2) * HID; i += 128) {
            const int j = i >> 7;            // row pair
            const int c = i & 127;           // column
            v2f w; w.x = W1[(2 * j) * HID + c]; w.y = W1[(2 * j + 1) * HID + c];
            *(v2f*)&sW1p[wpair_addr(j, c)] = w;
        }
        for (int i = tid; i < (HID / 2) * HID; i += 128) {
            const int j = i >> 7;
            const int c = i & 127;
            v2f w2; w2.x = W2[(2 * j) * HID + c]; w2.y = W2[(2 * j + 1) * HID + c];
            *(v2f*)&sW2p[wpair_addr(j, c)] = w2;
            v2f w3; w3.x = W3[(2 * j) * HID + c]; w3.y = W3[(2 * j + 1) * HID + c];
            *(v2f*)&sW3p[wpair_addr(j, c)] = w3;
        }
    }
    __syncthreads();   // all 128 threads reach this (no prior divergence)

    const int tile = blockIdx.x * WAVES_PER_BLOCK + wave;
    if (tile < N_TILES) {            // wave-uniform guard -> EXEC all-1s below
        const int r0 = tile * 16;
        float* hbuf = hbuf_all[wave];

        // Prefetch the residual row for this lane (global_prefetch_b8)
        __builtin_prefetch(nf + (size_t)(r0 + mcol) * FDIM, 0, 1);

        const int bsw = hh << 5;     // B-fragment XOR swizzle for this lane

        v8f acc[8];
        #pragma unroll
        for (int n = 0; n < 8; ++n) acc[n] = v8f{0.f,0.f,0.f,0.f,0.f,0.f,0.f,0.f};

        // ---- Layer 1: X[16 x 256] @ W1[256 x 128], X = concat(nf, agg) ----
        {
            const float* arow_nf = nf  + (size_t)(r0 + mcol) * FDIM;
            const float* arow_ag = agg + (size_t)(r0 + mcol) * FDIM;
            for (int k = 0; k < 64; ++k) {           // K = 256, 4 per WMMA
                const int c0 = k * 4 + hh * 2;       // row pair j = 2k + hh
                const float* asrc = (c0 < FDIM) ? (arow_nf + c0)
                                                : (arow_ag + (c0 - FDIM));
                const v2f a = *(const v2f*)asrc;
                const float* bbase = sW1p + (2 * k + hh) * 256;
                #pragma unroll
                for (int n = 0; n < 8; ++n) {
                    const v2f b = *(const v2f*)&bbase[((n * 16 + mcol) << 1) ^ bsw];
                    acc[n] = __builtin_amdgcn_wmma_f32_16x16x4_f32(
                        false, a, false, b, (short)0, acc[n], false, false);
                }
            }
            #pragma unroll
            for (int n = 0; n < 8; ++n) {            // bias + relu -> LDS
                const float bias = b1[n * 16 + mcol];
                #pragma unroll
                for (int v = 0; v < 8; ++v) {
                    float x = acc[n][v] + bias;
                    x = x > 0.f ? x : 0.f;
                    hbuf[(v + 8 * hh) * 130 + n * 16 + mcol] = x;
                }
            }
            __asm__ volatile("" ::: "memory");  // order LDS store -> LDS load
        }

        // ---- Layer 2: H[16 x 128] @ W2[128 x 128], relu -------------------
        {
            #pragma unroll
            for (int n = 0; n < 8; ++n) acc[n] = v8f{0.f,0.f,0.f,0.f,0.f,0.f,0.f,0.f};
            for (int k = 0; k < 32; ++k) {           // K = 128
                const int c0 = k * 4 + hh * 2;
                const v2f a = *(const v2f*)&hbuf[mcol * 130 + c0];
                const float* bbase = sW2p + (2 * k + hh) * 256;
                #pragma unroll
                for (int n = 0; n < 8; ++n) {
                    const v2f b = *(const v2f*)&bbase[((n * 16 + mcol) << 1) ^ bsw];
                    acc[n] = __builtin_amdgcn_wmma_f32_16x16x4_f32(
                        false, a, false, b, (short)0, acc[n], false, false);
                }
            }
            __asm__ volatile("" ::: "memory");  // all layer-2 A reads done
            #pragma unroll
            for (int n = 0; n < 8; ++n) {
                const float bias = b2[n * 16 + mcol];
                #pragma unroll
                for (int v = 0; v < 8; ++v) {
                    float x = acc[n][v] + bias;
                    x = x > 0.f ? x : 0.f;
                    hbuf[(v + 8 * hh) * 130 + n * 16 + mcol] = x;
                }
            }
            __asm__ volatile("" ::: "memory");
        }

        // ---- Layer 3: H[16 x 128] @ W3[128 x 128] + b3 --------------------
        {
            #pragma unroll
            for (int n = 0; n < 8; ++n) acc[n] = v8f{0.f,0.f,0.f,0.f,0.f,0.f,0.f,0.f};
            for (int k = 0; k < 32; ++k) {
                const int c0 = k * 4 + hh * 2;
                const v2f a = *(const v2f*)&hbuf[mcol * 130 + c0];
                const float* bbase = sW3p + (2 * k + hh) * 256;
                #pragma unroll
                for (int n = 0; n < 8; ++n) {
                    const v2f b = *(const v2f*)&bbase[((n * 16 + mcol) << 1) ^ bsw];
                    acc[n] = __builtin_amdgcn_wmma_f32_16x16x4_f32(
                        false, a, false, b, (short)0, acc[n], false, false);
                }
            }
            #pragma unroll
            for (int n = 0; n < 8; ++n) {
                const float bias = b3[n * 16 + mcol];
                #pragma unroll
                for (int v = 0; v < 8; ++v) acc[n][v] += bias;
            }
        }

        // ---- LayerNorm + residual ----------------------------------------
        // Lane holds rows {8*hh + v}, cols {n*16 + mcol}; reduce across the
        // 16 lanes of this half-wave (xor masks < 16 stay within the half).
        float mu[8], rs[8];
        #pragma unroll
        for (int v = 0; v < 8; ++v) {
            float s = 0.f, q = 0.f;
            #pragma unroll
            for (int n = 0; n < 8; ++n) {
                const float x = acc[n][v];
                s += x;
                q += x * x;
            }
            #pragma unroll
            for (int m = 1; m < 16; m <<= 1) {
                s += __shfl_xor(s, m, 32);
                q += __shfl_xor(q, m, 32);
            }
            const float mean = s * (1.0f / FDIM);
            const float var  = q * (1.0f / FDIM) - mean * mean;
            mu[v] = mean;
            rs[v] = rsqrtf(var + LN_EPS);
        }

        #pragma unroll
        for (int n = 0; n < 8; ++n) {
            const int col = n * 16 + mcol;
            const float g  = gamma[col];
            const float bt = beta[col];
            #pragma unroll
            for (int v = 0; v < 8; ++v) {
                const int row = r0 + v + 8 * hh;
                const size_t idx = (size_t)row * FDIM + col;
                out[idx] = (acc[n][v] - mu[v]) * rs[v] * g + bt + nf[idx];
            }
        }
    }
}

// ---------------------------------------------------------------------------
extern "C" void kernel_launch(void* const* d_in, const int* in_sizes, int n_in,
                              void* d_out, int out_size, void* d_ws, size_t ws_size,
                              hipStream_t stream) {
    const float* nf    = (const float*)d_in[0];
    const int*   em    = (const int*)  d_in[1];
    const float* ef    = (const float*)d_in[2];
    const float* W1    = (const float*)d_in[3];
    const float* b1    = (const float*)d_in[4];
    const float* W2    = (const float*)d_in[5];
    const float* b2    = (const float*)d_in[6];
    const float* W3    = (const float*)d_in[7];
    const float* b3    = (const float*)d_in[8];
    const float* gamma = (const float*)d_in[9];
    const float* beta  = (const float*)d_in[10];
    float* out = (float*)d_out;
    float* agg = (float*)d_ws;                   // N_NODE*FDIM floats (25.6 MB)

    (void)in_sizes; (void)n_in; (void)out_size; (void)ws_size;

    // Phase 0: zero agg (1,600,000 float4; 6250 * 256 exact)
    zero_agg_kernel<<<6250, 256, 0, stream>>>((float4*)agg);
    // Phase 1: scatter-add (20,000,000 threads; 78125 * 256 exact)
    scatter_add_kernel<<<78125, 256, 0, stream>>>((const float4*)ef, em, agg);
    // Phase 2: MLP + LN + residual (3125 tiles, 4 waves/block)
    const int blocks = (N_TILES + WAVES_PER_BLOCK - 1) / WAVES_PER_BLOCK; // 782
    mlp_ln_kernel<<<blocks, 32 * WAVES_PER_BLOCK, 0, stream>>>(
        nf, agg, W1, b1, W2, b2, W3, b3, gamma, beta, out);
}